// RelationUnit_70987219469020
// MI455X (gfx1250) — compile-verified
//
#include <hip/hip_runtime.h>
#include <cstdint>
#include <cstddef>

// ---------------------------------------------------------------------------
// RelationUnit on MI455X (gfx1250).
// HBM-bound: position_embedding is 1.07 GB fp32 (floor ~46us @ 23.3 TB/s);
// all GEMMs are tiny and L2-resident. Plan:
//   K1: wk/wq/wv projections via V_WMMA_F32_16X16X4_F32
//   K2: scores S = wk @ wq^T / 8 via the same WMMA (16 MB -> ws, L2-resident)
//   K3: fused per-row  log(max(pe.wg+b,1e-6)) + S  -> softmax -> p @ V.
//       pe tiles are streamed Global->LDS by the Tensor Data Mover
//       (tensor_load_to_lds + TENSORcnt), double-buffered so DMA of tile t+1
//       overlaps compute on tile t. TDM pad_interval/amount gives the 65-float
//       LDS row pitch (bank-conflict-free) in hardware.
// ---------------------------------------------------------------------------

typedef __attribute__((ext_vector_type(2))) float    v2f;
typedef __attribute__((ext_vector_type(8))) float    v8f;
typedef __attribute__((ext_vector_type(4))) uint32_t u32x4;
typedef __attribute__((ext_vector_type(8))) uint32_t u32x8;

#define NN 2048
#define DA 256
#define DK 64
#define TROWS 256                 // pe rows per tile (256*64 floats = 64 KB)
#define TPITCH 65                 // LDS row pitch after TDM padding

// ---------------- Kernel 1: projections  out = f_a @ W^T + b ----------------
__global__ __launch_bounds__(32) void proj_kernel(
    const float* __restrict__ fa,
    const float* __restrict__ Wk, const float* __restrict__ bk,
    const float* __restrict__ Wq, const float* __restrict__ bq,
    const float* __restrict__ Wv, const float* __restrict__ bv,
    float* __restrict__ ok, float* __restrict__ oq, float* __restrict__ ov)
{
    const float* W; const float* b; float* out;
    switch (blockIdx.z) {
        case 0:  W = Wk; b = bk; out = ok; break;
        case 1:  W = Wq; b = bq; out = oq; break;
        default: W = Wv; b = bv; out = ov; break;
    }
    const int lane = threadIdx.x;
    const int half = lane >> 4;         // 0: K pair {0,1}, 1: K pair {2,3}
    const int l    = lane & 15;
    const int n0 = blockIdx.x * 16;
    const int m0 = blockIdx.y * 16;

    v8f c = {};
    const float* arow = fa + (size_t)(m0 + l) * DA + 2 * half;   // A[M=l, K]
    const float* brow = W  + (size_t)(n0 + l) * DA + 2 * half;   // B[K,N=l]=W[l,K]
    for (int kk = 0; kk < DA; kk += 4) {
        v2f a, bb;
        a.x  = arow[kk];  a.y  = arow[kk + 1];
        bb.x = brow[kk];  bb.y = brow[kk + 1];
        c = __builtin_amdgcn_wmma_f32_16x16x4_f32(false, a, false, bb,
                                                  (short)0, c, false, false);
    }
    const float bias = b[n0 + l];
    #pragma unroll
    for (int r = 0; r < 8; ++r) {
        out[(size_t)(m0 + r + 8 * half) * DK + n0 + l] = c[r] + bias;
    }
}

// ---------------- Kernel 2: scores  S = (wk @ wq^T) * 0.125 -----------------
__global__ __launch_bounds__(32) void score_kernel(
    const float* __restrict__ wk, const float* __restrict__ wq,
    float* __restrict__ S)
{
    const int lane = threadIdx.x;
    const int half = lane >> 4;
    const int l    = lane & 15;
    const int n0 = blockIdx.x * 16;
    const int m0 = blockIdx.y * 16;

    v8f c = {};
    const float* arow = wk + (size_t)(m0 + l) * DK + 2 * half;
    const float* brow = wq + (size_t)(n0 + l) * DK + 2 * half;
    for (int kk = 0; kk < DK; kk += 4) {
        v2f a, bb;
        a.x  = arow[kk];  a.y  = arow[kk + 1];
        bb.x = brow[kk];  bb.y = brow[kk + 1];
        c = __builtin_amdgcn_wmma_f32_16x16x4_f32(false, a, false, bb,
                                                  (short)0, c, false, false);
    }
    #pragma unroll
    for (int r = 0; r < 8; ++r) {
        S[(size_t)(m0 + r + 8 * half) * NN + n0 + l] = c[r] * 0.125f;
    }
}

// ---- TDM: DMA one contiguous 64KB pe slab (256 rows x 64 f32) into LDS ----
// 1D tile of 16384 4-byte elements; pad 1 DWORD every 64 DWORDs -> pitch 65.
// D# group0: count=1, lds_addr[63:32], global_addr[120:64], type=2.
// D# group1: data_size=2(4B), pad_enable, pad_interval=5(64 dw), pad_amount=0,
//            tensor_dim0=tile_dim0=16384, tensor_dim1=tile_dim1=1.
__device__ __forceinline__ void tdm_load_tile(const void* gsrc, uint32_t lds_addr)
{
    const uint64_t ga = (uint64_t)(uintptr_t)gsrc;
    u32x4 g0;
    g0.s0 = 1u;                                            // count=1 (valid D#)
    g0.s1 = lds_addr;                                      // lds_addr[63:32]
    g0.s2 = (uint32_t)ga;                                  // global_addr[95:64]
    g0.s3 = ((uint32_t)(ga >> 32) & 0x01FFFFFFu) | (2u << 30);  // [120:96]|type=2

    u32x8 g1;
    g1.s0 = (2u << 16) | (1u << 20) | (5u << 22);          // 4B, pad, 64dw/1dw
    g1.s1 = (16384u << 16);                                // tensor_dim0[15:0]
    g1.s2 = (1u << 16);                                    // dim0 hi=0, dim1=1
    g1.s3 = (16384u << 16);                                // dim1 hi=0, tile_dim0
    g1.s4 = 1u;                                            // tile_dim1=1, dim2=0
    g1.s5 = 16384u;                                        // dim0_stride lo
    g1.s6 = (16384u << 16);                                // stride0 hi=0, s1 lo
    g1.s7 = 0u;                                            // dim1_stride hi
    asm volatile("tensor_load_to_lds %0, %1" :: "s"(g0), "s"(g1) : "memory");
}

// ---------------- Kernel 3: fused geo-bias + softmax + p @ V ----------------
__global__ __launch_bounds__(256) void attn_kernel(
    const float* __restrict__ pe,
    const float* __restrict__ wg, const float* __restrict__ wgb,
    const float* __restrict__ wk, const float* __restrict__ wq,
    const float* __restrict__ wv, const float* __restrict__ S,
    float* __restrict__ out)
{
    __shared__ float sh_tile[2][TROWS * TPITCH];  // 2 x 65 KB ping-pong
    __shared__ float sh_logit[NN];                //  8 KB
    __shared__ float sh_wg[64];
    __shared__ float sh_km[64];
    __shared__ float sh_red[256];
    __shared__ float sh_acc[256];

    const int tid = threadIdx.x;
    const int m   = blockIdx.x;
    const bool wave0 = (tid < 32);                // wave32: first wave issues DMA

    if (tid < 64) {
        sh_wg[tid] = wg[tid];
        sh_km[tid] = wk[(size_t)m * DK + tid];
    }
    const float gb = wgb[0];

    const char* peRow = (const char*)pe + (size_t)m * NN * DK * sizeof(float);
    const uint32_t ldsA = (uint32_t)(uintptr_t)&sh_tile[0][0];
    const uint32_t ldsB = (uint32_t)(uintptr_t)&sh_tile[1][0];

    if (wave0) tdm_load_tile(peRow, ldsA);        // prime the pipeline (tile 0)
    __syncthreads();                              // covers sh_wg/sh_km too

    // ---- Phase A: logits over n, double-buffered TDM tiles ----
    for (int t = 0; t < NN / TROWS; ++t) {
        if (wave0) {
            if (t + 1 < NN / TROWS) {
                tdm_load_tile(peRow + (size_t)(t + 1) * TROWS * DK * sizeof(float),
                              (t & 1) ? ldsA : ldsB);     // buf[(t+1)&1]
                __builtin_amdgcn_s_wait_tensorcnt(1);     // tile t complete
            } else {
                __builtin_amdgcn_s_wait_tensorcnt(0);     // last tile complete
            }
        }
        __syncthreads();                          // release all waves on tile t

        const float* trow = &sh_tile[t & 1][tid * TPITCH];
        float g = gb;
        #pragma unroll
        for (int j = 0; j < 64; ++j) g += trow[j] * sh_wg[j];  // conflict-free

        const int n = t * TROWS + tid;
        float s;
        if (S) {
            s = S[(size_t)m * NN + n];
        } else {
            const float* qn = wq + (size_t)n * DK;
            float d = 0.f;
            #pragma unroll
            for (int j = 0; j < 64; ++j) d += sh_km[j] * qn[j];
            s = d * 0.125f;
        }
        // relu then clip(1e-6) then log  ==  log(max(x, 1e-6))
        sh_logit[n] = __logf(fmaxf(g, 1e-6f)) + s;
        __syncthreads();                          // tile buf reusable by DMA
    }

    // ---- Phase B: softmax over n (max + sum trees in LDS) ----
    float lmax = -3.402823e38f;
    for (int n = tid; n < NN; n += 256) lmax = fmaxf(lmax, sh_logit[n]);
    sh_red[tid] = lmax;
    __syncthreads();
    for (int s2 = 128; s2 > 0; s2 >>= 1) {
        if (tid < s2) sh_red[tid] = fmaxf(sh_red[tid], sh_red[tid + s2]);
        __syncthreads();
    }
    const float rowmax = sh_red[0];
    __syncthreads();

    float lsum = 0.f;
    for (int n = tid; n < NN; n += 256) {
        const float e = __expf(sh_logit[n] - rowmax);
        sh_logit[n] = e;
        lsum += e;
    }
    sh_red[tid] = lsum;
    __syncthreads();
    for (int s2 = 128; s2 > 0; s2 >>= 1) {
        if (tid < s2) sh_red[tid] += sh_red[tid + s2];
        __syncthreads();
    }
    const float invsum = 1.0f / sh_red[0];
    __syncthreads();

    // ---- Phase C: out[m,k] = sum_n p[n] * wv[n,k] ----
    const int k  = tid & 63;
    const int ch = tid >> 6;                      // 4 chunks of 512 n each
    float partial = 0.f;
    for (int n = ch * 512; n < (ch + 1) * 512; ++n) {
        partial += sh_logit[n] * wv[(size_t)n * DK + k];   // coalesced across k
    }
    sh_acc[tid] = partial;
    __syncthreads();
    if (tid < 64) {
        const float tot = sh_acc[tid] + sh_acc[tid + 64] +
                          sh_acc[tid + 128] + sh_acc[tid + 192];
        out[(size_t)m * DK + tid] = tot * invsum;
    }
}

// ---------------------------------------------------------------------------
extern "C" void kernel_launch(void* const* d_in, const int* in_sizes, int n_in,
                              void* d_out, int out_size, void* d_ws, size_t ws_size,
                              hipStream_t stream) {
    (void)in_sizes; (void)n_in; (void)out_size;
    const float* fa  = (const float*)d_in[0];
    const float* pe  = (const float*)d_in[1];
    const float* WGw = (const float*)d_in[2];
    const float* WGb = (const float*)d_in[3];
    const float* WKw = (const float*)d_in[4];
    const float* WKb = (const float*)d_in[5];
    const float* WQw = (const float*)d_in[6];
    const float* WQb = (const float*)d_in[7];
    const float* WVw = (const float*)d_in[8];
    const float* WVb = (const float*)d_in[9];
    float* out = (float*)d_out;
    float* ws  = (float*)d_ws;

    const size_t projElems = (size_t)NN * DK;
    float* wk = ws;
    float* wq = ws + projElems;
    float* wv = ws + 2 * projElems;
    const size_t needS = (3 * projElems + (size_t)NN * NN) * sizeof(float);
    float* S = (ws_size >= needS) ? (ws + 3 * projElems) : nullptr;

    dim3 pg(DK / 16, NN / 16, 3);
    proj_kernel<<<pg, 32, 0, stream>>>(fa, WKw, WKb, WQw, WQb, WVw, WVb,
                                       wk, wq, wv);
    if (S) {
        dim3 sg(NN / 16, NN / 16);
        score_kernel<<<sg, 32, 0, stream>>>(wk, wq, S);
    }
    attn_kernel<<<NN, 256, 0, stream>>>(pe, WGw, WGb, wk, wq, wv, S, out);
}